// SupplyChainGNN_1906965479656
// MI455X (gfx1250) — compile-verified
//
#include <hip/hip_runtime.h>

typedef float v2f __attribute__((ext_vector_type(2)));
typedef float v8f __attribute__((ext_vector_type(8)));

// Only check the target builtin during the device (gfx1250) pass; it is
// naturally unavailable in the host x86 pass.
#if defined(__HIP_DEVICE_COMPILE__) && !__has_builtin(__builtin_amdgcn_wmma_f32_16x16x4_f32)
#error "wmma_f32_16x16x4_f32 builtin not available on gfx1250 device pass"
#endif

#define WG_THREADS 256
#define WG_WAVES 8
#define TILE_ROWS 128          // 8 waves * 16 rows
#define B_STRIDE 68            // padded LDS row stride for transposed B

// ---------------------------------------------------------------------------
// init: zero agg accumulator and degree counters
__global__ void k_init(float* __restrict__ agg, unsigned* __restrict__ deg,
                       int n64, int n) {
    int i = blockIdx.x * blockDim.x + threadIdx.x;
    if (i < n64) agg[i] = 0.0f;
    if (i < n)   deg[i] = 0u;
}

// degree: deg[dst] += 1 (exact integer atomics)
__global__ void k_degree(const int* __restrict__ ei, unsigned* __restrict__ deg,
                         int nE) {
    int e = blockIdx.x * blockDim.x + threadIdx.x;
    if (e >= nE) return;
    atomicAdd(&deg[ei[nE + e]], 1u);
}

// dinv = rsqrt(deg + 1)
__global__ void k_dinv(const unsigned* __restrict__ deg, float* __restrict__ dinv,
                       int n) {
    int i = blockIdx.x * blockDim.x + threadIdx.x;
    if (i >= n) return;
    dinv[i] = rsqrtf((float)deg[i] + 1.0f);
}

// node encoder: h = relu(x @ Wn + bn), x:[N,5] Wn:[5,64]
__global__ void k_encode(const float* __restrict__ x, const float* __restrict__ Wn,
                         const float* __restrict__ bn, float* __restrict__ h,
                         int n64) {
    int i = blockIdx.x * blockDim.x + threadIdx.x;
    if (i >= n64) return;
    int n = i >> 6, j = i & 63;
    float s = bn[j];
#pragma unroll
    for (int k = 0; k < 5; ++k) s = fmaf(x[n * 5 + k], Wn[k * 64 + j], s);
    h[i] = fmaxf(s, 0.0f);
}

// ---------------------------------------------------------------------------
// hw = h @ W   (W is 64x64), f32 WMMA 16x16x4, LDS-staged tiles.
__global__ __launch_bounds__(WG_THREADS)
void k_gemm64(const float* __restrict__ h, const float* __restrict__ W,
              float* __restrict__ hw, int nrows) {
    __shared__ float sA[TILE_ROWS * 64];      // h tile, row-major
    __shared__ float sBT[64 * B_STRIDE];      // W transposed: sBT[n*68+k] = W[k*64+n]

    const int t    = threadIdx.x;
    const int wave = t >> 5;
    const int lane = t & 31;
    const int half = lane >> 4;               // K split: lanes 16-31 hold K+2,K+3
    const int m    = lane & 15;               // A: row; B/C/D: column
    const int rowBase = blockIdx.x * TILE_ROWS;

    // stage W transposed (coalesced read, one-time strided LDS write)
    for (int i = t; i < 64 * 64; i += WG_THREADS) {
        int k = i >> 6, n = i & 63;
        sBT[n * B_STRIDE + k] = W[i];
    }
    // stage 128x64 A tile (float4 coalesced, zero-fill OOB rows)
    for (int i = t * 4; i < TILE_ROWS * 64; i += WG_THREADS * 4) {
        int row = rowBase + (i >> 6);
        float4 v = make_float4(0.f, 0.f, 0.f, 0.f);
        if (row < nrows) v = *(const float4*)(h + row * 64 + (i & 63));
        *(float4*)(sA + i) = v;
    }
    __syncthreads();

    v8f acc0 = {}, acc1 = {}, acc2 = {}, acc3 = {};
    const float* aRow = sA + (wave * 16 + m) * 64;

#pragma unroll
    for (int s = 0; s < 16; ++s) {
        const int k0 = 4 * s + 2 * half;
        // A 16x4 f32 fragment: lane(m,half) holds A[m][k0], A[m][k0+1]
        v2f a = *(const v2f*)(aRow + k0);
        // B 4x16 f32 fragments: lane(m,half) holds B[k0][m], B[k0+1][m]
        v2f b0 = *(const v2f*)(sBT + (0 * 16 + m) * B_STRIDE + k0);
        acc0 = __builtin_amdgcn_wmma_f32_16x16x4_f32(false, a, false, b0, (short)0, acc0, false, false);
        v2f b1 = *(const v2f*)(sBT + (1 * 16 + m) * B_STRIDE + k0);
        acc1 = __builtin_amdgcn_wmma_f32_16x16x4_f32(false, a, false, b1, (short)0, acc1, false, false);
        v2f b2 = *(const v2f*)(sBT + (2 * 16 + m) * B_STRIDE + k0);
        acc2 = __builtin_amdgcn_wmma_f32_16x16x4_f32(false, a, false, b2, (short)0, acc2, false, false);
        v2f b3 = *(const v2f*)(sBT + (3 * 16 + m) * B_STRIDE + k0);
        acc3 = __builtin_amdgcn_wmma_f32_16x16x4_f32(false, a, false, b3, (short)0, acc3, false, false);
    }

    // C/D layout: VGPR r -> row (r + 8*half), col = m (+16 per N-tile)
    const int myRow0 = rowBase + wave * 16;
#pragma unroll
    for (int r = 0; r < 8; ++r) {
        int row = myRow0 + r + 8 * half;
        if (row < nrows) {
            float* o = hw + row * 64 + m;
            o[0]  = acc0[r];
            o[16] = acc1[r];
            o[32] = acc2[r];
            o[48] = acc3[r];
        }
    }
}

// ---------------------------------------------------------------------------
// scatter: one wave per edge; agg[dst] += norm_e * hw[src]
__global__ void k_scatter(const int* __restrict__ ei, const float* __restrict__ hw,
                          const float* __restrict__ dinv, float* __restrict__ agg,
                          int nE) {
    int w    = (blockIdx.x * blockDim.x + threadIdx.x) >> 5;
    int lane = threadIdx.x & 31;
    if (w >= nE) return;
    int src = ei[w];
    int dst = ei[nE + w];
    float nrm = dinv[src] * dinv[dst];
    float v0 = hw[src * 64 + lane]      * nrm;
    float v1 = hw[src * 64 + 32 + lane] * nrm;
    unsafeAtomicAdd(&agg[dst * 64 + lane],      v0);
    unsafeAtomicAdd(&agg[dst * 64 + 32 + lane], v1);
}

// combine: h = relu(agg + dinv^2 * hw + bc); reset agg for next layer
__global__ void k_combine(const float* __restrict__ dinv, const float* __restrict__ hw,
                          const float* __restrict__ bc, float* __restrict__ agg,
                          float* __restrict__ h, int n64) {
    int i = blockIdx.x * blockDim.x + threadIdx.x;
    if (i >= n64) return;
    int n = i >> 6, j = i & 63;
    float d = dinv[n];
    float v = agg[i] + d * d * hw[i] + bc[j];
    h[i]   = fmaxf(v, 0.0f);
    agg[i] = 0.0f;
}

// ---------------------------------------------------------------------------
// heads: one wave per node; lane = hidden unit (H/2 = 32 = wave32)
__global__ __launch_bounds__(256)
void k_heads(const float* __restrict__ h,
             const float* __restrict__ Wd1, const float* __restrict__ bd1,
             const float* __restrict__ Wd2, const float* __restrict__ bd2,
             const float* __restrict__ Wi1, const float* __restrict__ bi1,
             const float* __restrict__ Wi2, const float* __restrict__ bi2,
             float* __restrict__ out, int n_nodes) {
    int w    = (blockIdx.x * blockDim.x + threadIdx.x) >> 5;
    int lane = threadIdx.x & 31;
    if (w >= n_nodes) return;
    float hlo = h[w * 64 + lane];        // coalesced; broadcast via shfl
    float hhi = h[w * 64 + 32 + lane];
    float sd = bd1[lane], si = bi1[lane];
#pragma unroll
    for (int k = 0; k < 32; ++k) {
        float hv = __shfl(hlo, k, 32);
        sd = fmaf(hv, Wd1[k * 32 + lane], sd);
        si = fmaf(hv, Wi1[k * 32 + lane], si);
    }
#pragma unroll
    for (int k = 0; k < 32; ++k) {
        float hv = __shfl(hhi, k, 32);
        sd = fmaf(hv, Wd1[(k + 32) * 32 + lane], sd);
        si = fmaf(hv, Wi1[(k + 32) * 32 + lane], si);
    }
    sd = fmaxf(sd, 0.0f) * Wd2[lane];
    si = fmaxf(si, 0.0f) * Wi2[lane];
#pragma unroll
    for (int off = 16; off > 0; off >>= 1) {
        sd += __shfl_xor(sd, off, 32);
        si += __shfl_xor(si, off, 32);
    }
    if (lane == 0) {
        out[w]           = sd + bd2[0];
        out[n_nodes + w] = si + bi2[0];
    }
}

// ---------------------------------------------------------------------------
static inline size_t alignUp(size_t v) { return (v + 255) & ~size_t(255); }

extern "C" void kernel_launch(void* const* d_in, const int* in_sizes, int n_in,
                              void* d_out, int out_size, void* d_ws, size_t ws_size,
                              hipStream_t stream) {
    const float* x   = (const float*)d_in[0];
    const int*   ei  = (const int*)  d_in[1];
    // d_in[2] edge_attr, d_in[5] We, d_in[6] be: dead code in the reference
    const float* Wn  = (const float*)d_in[3];
    const float* bn  = (const float*)d_in[4];
    const float* Wc  = (const float*)d_in[7];
    const float* bc  = (const float*)d_in[8];
    const float* Wd1 = (const float*)d_in[9];
    const float* bd1 = (const float*)d_in[10];
    const float* Wd2 = (const float*)d_in[11];
    const float* bd2 = (const float*)d_in[12];
    const float* Wi1 = (const float*)d_in[13];
    const float* bi1 = (const float*)d_in[14];
    const float* Wi2 = (const float*)d_in[15];
    const float* bi2 = (const float*)d_in[16];
    float* out = (float*)d_out;

    const int N = in_sizes[0] / 5;         // 100000
    const int E = in_sizes[1] / 2;         // 1600000
    const int L = in_sizes[7] / (64 * 64); // 3
    const int N64 = N * 64;

    // workspace carve-out (~78 MB)
    char* p = (char*)d_ws;
    float*    h    = (float*)p;    p += alignUp((size_t)N64 * 4);
    float*    hw   = (float*)p;    p += alignUp((size_t)N64 * 4);
    float*    agg  = (float*)p;    p += alignUp((size_t)N64 * 4);
    unsigned* deg  = (unsigned*)p; p += alignUp((size_t)N * 4);
    float*    dinv = (float*)p;

    const int T = 256;
    k_init  <<<(N64 + T - 1) / T, T, 0, stream>>>(agg, deg, N64, N);
    k_degree<<<(E + T - 1) / T,   T, 0, stream>>>(ei, deg, E);
    k_dinv  <<<(N + T - 1) / T,   T, 0, stream>>>(deg, dinv, N);
    k_encode<<<(N64 + T - 1) / T, T, 0, stream>>>(x, Wn, bn, h, N64);

    const int gemmBlocks    = (N + TILE_ROWS - 1) / TILE_ROWS;
    const int scatterBlocks = (E * 32 + T - 1) / T;   // one wave per edge
    for (int l = 0; l < L; ++l) {
        k_gemm64 <<<gemmBlocks, WG_THREADS, 0, stream>>>(h, Wc + l * 64 * 64, hw, N);
        k_scatter<<<scatterBlocks, T, 0, stream>>>(ei, hw, dinv, agg, E);
        k_combine<<<(N64 + T - 1) / T, T, 0, stream>>>(dinv, hw, bc + l * 64, agg, h, N64);
    }

    k_heads<<<(N * 32 + T - 1) / T, T, 0, stream>>>(h, Wd1, bd1, Wd2, bd2,
                                                    Wi1, bi1, Wi2, bi2, out, N);
}